// SelectiveSSM_28948079575407
// MI455X (gfx1250) — compile-verified
//
#include <hip/hip_runtime.h>
#include <hip/hip_bf16.h>

// ---------------------------------------------------------------------------
// Mamba selective-SSM block for MI455X (gfx1250, wave32, WMMA).
// GEMMs use V_WMMA_F32_16X16X4_F32 (fp32-exact matrix path, 2048 FLOP/instr),
// double-buffered LDS staging, and GLOBAL_LOAD_ASYNC_TO_LDS_B128 when the
// toolchain exposes it (ASYNCcnt path), else sync loads + ds_store.
// ---------------------------------------------------------------------------

typedef float v2f __attribute__((ext_vector_type(2)));
typedef float v8f __attribute__((ext_vector_type(8)));
typedef int   v4i_gcc __attribute__((vector_size(16)));   // matches builtin param

#define BSZ      2
#define SEQ      2048
#define DMODEL   1024
#define DINNER   2048
#define DSTATE   16
#define DTRANK   64
#define NPROJ    (2 * DINNER + 2 * DSTATE + DTRANK)   // 4192
#define MROWS    (BSZ * SEQ)                          // 4096

// K-chunk 16, LDS row pitch 20 floats = 80 B (16B-aligned rows; 20*lane mod 64
// hits 16 distinct banks per half-wave -> conflict-free v2f fragment reads).
#define KC       16
#define LDP      20

#if defined(__has_builtin)
#if __has_builtin(__builtin_amdgcn_global_load_async_to_lds_b128)
#define ASYNC_LDS 1
#endif
#endif

__device__ __forceinline__ float silu_f(float x) {
    return x * (1.0f / (1.0f + __expf(-x)));
}

#ifdef ASYNC_LDS
typedef __attribute__((address_space(1))) v4i_gcc* gptr_t;
typedef __attribute__((address_space(3))) v4i_gcc* lptr_t;
__device__ __forceinline__ void async_cp16(const float* g, float* l) {
    __builtin_amdgcn_global_load_async_to_lds_b128(
        (gptr_t)g, (lptr_t)l, 0, 0);
}
__device__ __forceinline__ void wait_async0() {
#if __has_builtin(__builtin_amdgcn_s_wait_asynccnt)
    __builtin_amdgcn_s_wait_asynccnt(0);
#else
    asm volatile("s_wait_asynccnt 0x0" ::: "memory");
#endif
}
#endif

// ---------------------------------------------------------------------------
// Tiled fp32 WMMA GEMM:  C[M,N] = A[M,K] * B[N,K]^T  (+ epilogue)
//   Block: 256 thr = 8 waves; block tile 128x128; K double-buffered in LDS.
//   Wave (wm,wn): 2 M-tiles x 4 N-tiles (8 v8f accumulators).
//   EPI: 0 = plain store, 1 = softplus(acc + ep0[col]), 2 = acc + ep0[row,col]
// ---------------------------------------------------------------------------
template <int EPI>
__global__ __launch_bounds__(256)
void wmma_gemm_kernel(const float* __restrict__ A, int lda,
                      const float* __restrict__ B, int ldb,
                      float* __restrict__ C, int ldc,
                      int M, int N, int K,
                      const float* __restrict__ ep0) {
    __shared__ float As[2][128 * LDP];
    __shared__ float Bs[2][128 * LDP];

    const int t    = threadIdx.x;
    const int lane = t & 31;
    const int wave = t >> 5;
    const int lm   = lane & 15;
    const int half = lane >> 4;
    const int wm   = wave & 3;       // M wave group (32 rows)
    const int wn   = wave >> 2;      // N wave group (64 cols)
    const int row0 = blockIdx.y * 128;
    const int col0 = blockIdx.x * 128;

    v8f acc[2][4] = {};
    const int nchunks = K / KC;

    // Per-thread cooperative-load slots: tile 128 x 16 = 512 float4 slots.
#ifdef ASYNC_LDS
    auto issue = [&](int buf, int k0) {
#pragma unroll
        for (int i = 0; i < 2; ++i) {
            int slot = t + i * 256;
            int r = slot >> 2, c4 = (slot & 3) * 4;
            async_cp16(A + (size_t)(row0 + r) * lda + k0 + c4,
                       &As[buf][r * LDP + c4]);
            int gr = col0 + r; if (gr > N - 1) gr = N - 1;
            async_cp16(B + (size_t)gr * ldb + k0 + c4,
                       &Bs[buf][r * LDP + c4]);
        }
    };
#else
    float4 ra[2], rb[2];
    auto fetch = [&](int k0) {
#pragma unroll
        for (int i = 0; i < 2; ++i) {
            int slot = t + i * 256;
            int r = slot >> 2, c4 = (slot & 3) * 4;
            ra[i] = *reinterpret_cast<const float4*>(
                A + (size_t)(row0 + r) * lda + k0 + c4);
            int gr = col0 + r; if (gr > N - 1) gr = N - 1;
            rb[i] = *reinterpret_cast<const float4*>(
                B + (size_t)gr * ldb + k0 + c4);
        }
    };
    auto stash = [&](int buf) {
#pragma unroll
        for (int i = 0; i < 2; ++i) {
            int slot = t + i * 256;
            int r = slot >> 2, c4 = (slot & 3) * 4;
            *reinterpret_cast<float4*>(&As[buf][r * LDP + c4]) = ra[i];
            *reinterpret_cast<float4*>(&Bs[buf][r * LDP + c4]) = rb[i];
        }
    };
#endif

    auto compute = [&](int buf) {
#pragma unroll
        for (int kk = 0; kk < KC / 4; ++kk) {
            // A frag: lanes 0-15 hold K={0,1}, lanes 16-31 K={2,3} (ISA layout)
            v2f a0 = *reinterpret_cast<const v2f*>(
                &As[buf][(wm * 32 + lm) * LDP + kk * 4 + 2 * half]);
            v2f a1 = *reinterpret_cast<const v2f*>(
                &As[buf][(wm * 32 + 16 + lm) * LDP + kk * 4 + 2 * half]);
#pragma unroll
            for (int nt = 0; nt < 4; ++nt) {
                v2f b = *reinterpret_cast<const v2f*>(
                    &Bs[buf][(wn * 64 + nt * 16 + lm) * LDP + kk * 4 + 2 * half]);
                acc[0][nt] = __builtin_amdgcn_wmma_f32_16x16x4_f32(
                    false, a0, false, b, (short)0, acc[0][nt], false, false);
                acc[1][nt] = __builtin_amdgcn_wmma_f32_16x16x4_f32(
                    false, a1, false, b, (short)0, acc[1][nt], false, false);
            }
        }
    };

#ifdef ASYNC_LDS
    // Async pipeline: issue(c+1) overlaps compute(c); 1 barrier per chunk.
    issue(0, 0);
    for (int c = 0; c < nchunks; ++c) {
        wait_async0();
        __syncthreads();
        if (c + 1 < nchunks) issue((c + 1) & 1, (c + 1) * KC);
        compute(c & 1);
    }
#else
    // Sync pipeline: global->regs for (c+1) overlaps compute(c).
    fetch(0);
    stash(0);
    __syncthreads();
    for (int c = 0; c < nchunks; ++c) {
        if (c + 1 < nchunks) fetch((c + 1) * KC);
        compute(c & 1);
        if (c + 1 < nchunks) stash((c + 1) & 1);
        __syncthreads();
    }
#endif

    // Store D: vgpr e -> row = e + 8*half, col = lane&15 (ISA C/D layout).
#pragma unroll
    for (int mt = 0; mt < 2; ++mt) {
#pragma unroll
        for (int nt = 0; nt < 4; ++nt) {
            int col = col0 + wn * 64 + nt * 16 + lm;
            if (col >= N) continue;
#pragma unroll
            for (int e = 0; e < 8; ++e) {
                int   row = row0 + wm * 32 + mt * 16 + half * 8 + e;
                float val = acc[mt][nt][e];
                if (EPI == 1) {   // dt projection: softplus(acc + b_dt)
                    val += ep0[col];
                    val = (val > 20.0f) ? val : log1pf(__expf(val));
                }
                if (EPI == 2) {   // out projection: + residual
                    val += ep0[(size_t)row * ldc + col];
                }
                C[(size_t)row * ldc + col] = val;
            }
        }
    }
}

// ---------------------------------------------------------------------------
// Depthwise causal conv1d(k=4, left pad 3) + bias + SiLU.
// ---------------------------------------------------------------------------
__global__ __launch_bounds__(256)
void conv_silu_kernel(const float* __restrict__ proj,
                      const float* __restrict__ cw,
                      const float* __restrict__ cb,
                      float* __restrict__ xc) {
    int idx = blockIdx.x * 256 + threadIdx.x;     // over MROWS*DINNER
    int d = idx & (DINNER - 1);
    int m = idx >> 11;
    int l = m & (SEQ - 1);
    float acc = cb[d];
#pragma unroll
    for (int j = 0; j < 4; ++j) {
        int li = l - 3 + j;
        if (li >= 0)
            acc = fmaf(proj[(size_t)(m - 3 + j) * NPROJ + d], cw[d * 4 + j], acc);
    }
    xc[idx] = silu_f(acc);
}

// ---------------------------------------------------------------------------
// Selective scan: one thread per (b, d, n). 65536 threads = 2048 waves.
// h <- exp(dt*A_n)*h + dt*B_n*u ;  y = sum_n h*C_n  via 4 shfl_xor steps.
// ---------------------------------------------------------------------------
__global__ __launch_bounds__(256)
void scan_kernel(const float* __restrict__ proj,
                 const float* __restrict__ xc,
                 const float* __restrict__ dtb,
                 const float* __restrict__ A_log,
                 const float* __restrict__ Dvec,
                 float* __restrict__ y) {
    int t = blockIdx.x * 256 + threadIdx.x;
    int n = t & 15;
    int d = (t >> 4) & (DINNER - 1);
    int b = t >> 15;                                  // 16*2048 = 1<<15

    const float An = -__expf(A_log[d * DSTATE + n]);  // A = -exp(A_log)
    const float Dd = Dvec[d];
    float h = 0.0f;

    size_t base = (size_t)b * SEQ * DINNER + d;
    size_t prow = (size_t)b * SEQ * NPROJ;

    for (int l = 0; l < SEQ; ++l) {
        float dtv = dtb[base];
        float uv  = xc[base];
        float bn  = proj[prow + 2 * DINNER + n];
        float cn  = proj[prow + 2 * DINNER + DSTATE + n];
        float dA  = __expf(dtv * An);
        h = fmaf(dA, h, dtv * bn * uv);
        float p = h * cn;
        p += __shfl_xor(p, 1, 16);
        p += __shfl_xor(p, 2, 16);
        p += __shfl_xor(p, 4, 16);
        p += __shfl_xor(p, 8, 16);
        if (n == 0) y[base] = fmaf(uv, Dd, p);
        base += DINNER;
        prow += NPROJ;
    }
}

// ---------------------------------------------------------------------------
// y *= silu(z)   (z = proj[:, DINNER : 2*DINNER])
// ---------------------------------------------------------------------------
__global__ __launch_bounds__(256)
void gate_kernel(const float* __restrict__ proj, float* __restrict__ y) {
    int idx = blockIdx.x * 256 + threadIdx.x;
    size_t m = (size_t)(idx >> 11);
    int    c = idx & (DINNER - 1);
    float  z = proj[m * NPROJ + DINNER + c];
    y[idx] *= silu_f(z);
}

// ---------------------------------------------------------------------------
// LayerNorm over last dim (1024), in place on d_out. One block per row.
// ---------------------------------------------------------------------------
__global__ __launch_bounds__(256)
void ln_kernel(float* __restrict__ out,
               const float* __restrict__ g,
               const float* __restrict__ bia) {
    __shared__ float sA[8], sB[8];
    float* p = out + (size_t)blockIdx.x * DMODEL;
    float v[4], s = 0.0f, ss = 0.0f;
#pragma unroll
    for (int i = 0; i < 4; ++i) {
        v[i] = p[threadIdx.x + i * 256];
        s += v[i]; ss += v[i] * v[i];
    }
#pragma unroll
    for (int off = 16; off >= 1; off >>= 1) {
        s  += __shfl_xor(s,  off, 32);
        ss += __shfl_xor(ss, off, 32);
    }
    int wave = threadIdx.x >> 5, lane = threadIdx.x & 31;
    if (lane == 0) { sA[wave] = s; sB[wave] = ss; }
    __syncthreads();
    float st = 0.0f, sst = 0.0f;
#pragma unroll
    for (int i = 0; i < 8; ++i) { st += sA[i]; sst += sB[i]; }
    const float mu  = st * (1.0f / DMODEL);
    const float var = sst * (1.0f / DMODEL) - mu * mu;
    const float inv = rsqrtf(var + 1e-5f);
#pragma unroll
    for (int i = 0; i < 4; ++i) {
        int c = threadIdx.x + i * 256;
        p[c] = (v[i] - mu) * inv * g[c] + bia[c];
    }
}

// ---------------------------------------------------------------------------
extern "C" void kernel_launch(void* const* d_in, const int* in_sizes, int n_in,
                              void* d_out, int out_size, void* d_ws, size_t ws_size,
                              hipStream_t stream) {
    const float* x      = (const float*)d_in[0];
    const float* W_in   = (const float*)d_in[1];
    const float* W_dt   = (const float*)d_in[2];
    const float* b_dt   = (const float*)d_in[3];
    const float* A_log  = (const float*)d_in[4];
    const float* Dv     = (const float*)d_in[5];
    const float* conv_w = (const float*)d_in[6];
    const float* conv_b = (const float*)d_in[7];
    const float* W_out  = (const float*)d_in[8];
    const float* ln_g   = (const float*)d_in[9];
    const float* ln_b   = (const float*)d_in[10];

    float* ws   = (float*)d_ws;
    float* proj = ws;                                    // [4096, 4192]
    float* xc   = proj + (size_t)MROWS * NPROJ;          // [4096, 2048]
    float* dtb  = xc   + (size_t)MROWS * DINNER;         // [4096, 2048]
    float* y    = dtb  + (size_t)MROWS * DINNER;         // [4096, 2048]
    float* out  = (float*)d_out;                         // [4096, 1024]

    dim3 blk(256);

    // 1) in_proj: proj = x @ W_in^T    (M=4096, N=4192, K=1024)
    wmma_gemm_kernel<0><<<dim3((NPROJ + 127) / 128, MROWS / 128), blk, 0, stream>>>(
        x, DMODEL, W_in, DMODEL, proj, NPROJ, MROWS, NPROJ, DMODEL, nullptr);

    // 2) depthwise causal conv + SiLU on x_s slice of proj
    conv_silu_kernel<<<dim3(MROWS * DINNER / 256), blk, 0, stream>>>(
        proj, conv_w, conv_b, xc);

    // 3) dt = softplus(dt_raw @ W_dt^T + b_dt)   (M=4096, N=2048, K=64)
    wmma_gemm_kernel<1><<<dim3(DINNER / 128, MROWS / 128), blk, 0, stream>>>(
        proj + 2 * DINNER + 2 * DSTATE, NPROJ, W_dt, DTRANK, dtb, DINNER,
        MROWS, DINNER, DTRANK, b_dt);

    // 4) selective scan
    scan_kernel<<<dim3(BSZ * DINNER * DSTATE / 256), blk, 0, stream>>>(
        proj, xc, dtb, A_log, Dv, y);

    // 5) gate: y *= silu(z)
    gate_kernel<<<dim3(MROWS * DINNER / 256), blk, 0, stream>>>(proj, y);

    // 6) out = residual + y @ W_out^T   (M=4096, N=1024, K=2048)
    wmma_gemm_kernel<2><<<dim3(DMODEL / 128, MROWS / 128), blk, 0, stream>>>(
        y, DINNER, W_out, DINNER, out, DMODEL, MROWS, DMODEL, DINNER, x);

    // 7) LayerNorm in place on d_out
    ln_kernel<<<dim3(MROWS), blk, 0, stream>>>(out, ln_g, ln_b);
}